// Sampler_81527069213263
// MI455X (gfx1250) — compile-verified
//
#include <hip/hip_runtime.h>
#include <stdint.h>

// ---------------------------------------------------------------------------
// Sampler: per-row top-k (k<=100) + temperature + top-p + deterministic
// multinomial(0.5). Only the top-128 logits per row can affect the output
// (all other entries underflow to prob == 0.0f exactly), so we do an exact
// radix-style top-128 select per row, then a tiny finalize pass.
//
// Data movement (MI455X/gfx1250):
//   Phase 1 (HBM pass): TDM tensor_load_to_lds, leader-wave issue,
//                       double-buffered, TENSORcnt.
//   Phase 2 (L2 pass):  per-lane global_load_async_to_lds_b128, ASYNCcnt.
//   Phase 3 (L2 pass):  direct b128 loads + global_prefetch.
// ---------------------------------------------------------------------------

#define TOPN      128    // survivors kept per row (>= max top_k of 100)
#define NBINS     8192   // 13-bit histogram
#define CAND_CAP  1024   // candidate buffer (bitonic-sort size, power of 2)
#define T1        512    // threads for the select kernel
#define CHUNK_V4  512    // float4 per staged chunk (== T1) -> 8 KB
#define FT        128    // threads for finalize kernel

typedef uint32_t v4u __attribute__((ext_vector_type(4)));
typedef uint32_t v8u __attribute__((ext_vector_type(8)));

// Monotonic key: larger float -> larger unsigned key (no NaNs in input).
__device__ __forceinline__ uint32_t fkey(float x) {
  uint32_t u = __float_as_uint(x);
  return (u & 0x80000000u) ? ~u : (u | 0x80000000u);
}

// ---- CDNA5 per-lane async global->LDS copy (ASYNCcnt path) ----------------
// the LDS address comes from the VDST VGPR; the low 32 bits of a generic pointer to a
// __shared__ object are exactly the wave-relative LDS byte address.
__device__ __forceinline__ void async_copy_b128(const float4* gsrc, float4* lds_dst) {
  uint32_t lds_off = (uint32_t)(uintptr_t)(void*)lds_dst;
  uint64_t gaddr   = (uint64_t)(uintptr_t)(const void*)gsrc;
  asm volatile("global_load_async_to_lds_b128 %0, %1, off"
               :: "v"(lds_off), "v"(gaddr)
               : "memory");
}
__device__ __forceinline__ void wait_async_le(int n) {
  if (n == 0) asm volatile("s_wait_asynccnt 0x0" ::: "memory");
  else        asm volatile("s_wait_asynccnt 0x1" ::: "memory");
}

// ---- CDNA5 Tensor Data Mover: 1-D tile (nelem dwords) -> LDS --------------
// D# packing per CDNA5 ISA:
//   group0: count=1 | lds_addr[63:32] | global_addr[120:64] | type=2[127:126]
//   group1: data_size=4B, tensor_dim0=N, tensor_dim1=1, tile_dim0=N,
//           tile_dim1=1, tile_dim2=0, tensor_dim0_stride=N
//   groups 2/3: zero (unused dims).
__device__ __forceinline__ void tdm_load_1d(uint32_t lds_off, const void* gptr,
                                            uint32_t nelem) {
  uint64_t ga = (uint64_t)(uintptr_t)gptr;
  v4u g0;
  g0[0] = 1u;                                              // count=1
  g0[1] = lds_off;                                         // lds_addr
  g0[2] = (uint32_t)ga;                                    // global_addr[31:0]
  g0[3] = ((uint32_t)(ga >> 32) & 0x1FFFFFFu) | 0x80000000u; // [56:32] | type=2
  v8u g1;
  g1[0] = 0x00020000u;                                     // data_size=2 (4B)
  g1[1] = (nelem & 0xFFFFu) << 16;                         // tensor_dim0[15:0]
  g1[2] = (nelem >> 16) | (1u << 16);                      // dim0[31:16] | dim1=1
  g1[3] = (nelem & 0xFFFFu) << 16;                         // tile_dim0=N
  g1[4] = 1u;                                              // tile_dim1=1, tile_dim2=0
  g1[5] = nelem;                                           // dim0_stride[31:0]
  g1[6] = 0u;
  g1[7] = 0u;
  v4u gz = {0u, 0u, 0u, 0u};
  asm volatile("tensor_load_to_lds %0, %1, %2, %3"
               :: "s"(g0), "s"(g1), "s"(gz), "s"(gz)
               : "memory");
}
__device__ __forceinline__ void wait_tensor_le(int n) {
#if defined(__has_builtin) && __has_builtin(__builtin_amdgcn_s_wait_tensorcnt)
  // SIMM16 must be a literal constant -> branch to constant-arg calls.
  if (n == 0) __builtin_amdgcn_s_wait_tensorcnt(0);
  else        __builtin_amdgcn_s_wait_tensorcnt(1);
#else
  if (n == 0) asm volatile("s_wait_tensorcnt 0x0" ::: "memory");
  else        asm volatile("s_wait_tensorcnt 0x1" ::: "memory");
#endif
}

__global__ __launch_bounds__(T1) void topk_select_kernel(
    const float* __restrict__ logits, int V,
    float* __restrict__ topv, int* __restrict__ topi) {
  const int row = blockIdx.x;
  const int tid = threadIdx.x;
  const float* rowp = logits + (size_t)row * (size_t)V;
  const float4* rowp4 = (const float4*)rowp;

  __shared__ uint32_t hist[NBINS];              // 32 KB
  __shared__ float4   stage[2][CHUNK_V4];       // 16 KB staging (TDM / async)
  __shared__ float    cand_val[CAND_CAP];       // 4 KB
  __shared__ int      cand_idx[CAND_CAP];       // 4 KB
  __shared__ int      s_b1, s_cum1, s_count;
  __shared__ uint32_t s_thresh;

  const int nV4    = V >> 2;
  const int nChunk = nV4 / CHUNK_V4;
  // Scalar (SGPR) leader predicate -> scalar branch around TDM issue.
  const bool leader = (__builtin_amdgcn_readfirstlane((int)threadIdx.x) < 32);
  const uint32_t stage_off0 = (uint32_t)(uintptr_t)(void*)&stage[0][0];
  const uint32_t stage_off1 = (uint32_t)(uintptr_t)(void*)&stage[1][0];

  // ---------- Phase 1: coarse 13-bit histogram (HBM pass, TDM-staged) ------
  for (int i = tid; i < NBINS; i += T1) hist[i] = 0;
  __syncthreads();

  if (leader) {
    if (nChunk > 0) tdm_load_1d(stage_off0, rowp4, CHUNK_V4 * 4);
  }
  for (int c = 0; c < nChunk; ++c) {
    if (leader) {
      if (c + 1 < nChunk)
        tdm_load_1d(((c + 1) & 1) ? stage_off1 : stage_off0,
                    rowp4 + (size_t)(c + 1) * CHUNK_V4, CHUNK_V4 * 4);
      wait_tensor_le((c + 1 < nChunk) ? 1 : 0);  // chunk c done (TDM in-order)
    }
    __syncthreads();                             // LDS visible to all waves
    float4 v = stage[c & 1][tid];
    atomicAdd(&hist[fkey(v.x) >> 19], 1u);
    atomicAdd(&hist[fkey(v.y) >> 19], 1u);
    atomicAdd(&hist[fkey(v.z) >> 19], 1u);
    atomicAdd(&hist[fkey(v.w) >> 19], 1u);
    __syncthreads();                             // buffer reusable next chunk
  }
  // remainder float4s (direct loads) + scalar tail
  for (int i = nChunk * CHUNK_V4 + tid; i < nV4; i += T1) {
    float4 v = rowp4[i];
    atomicAdd(&hist[fkey(v.x) >> 19], 1u);
    atomicAdd(&hist[fkey(v.y) >> 19], 1u);
    atomicAdd(&hist[fkey(v.z) >> 19], 1u);
    atomicAdd(&hist[fkey(v.w) >> 19], 1u);
  }
  for (int i = (nV4 << 2) + tid; i < V; i += T1) {
    atomicAdd(&hist[fkey(rowp[i]) >> 19], 1u);
  }
  __syncthreads();

  if (tid == 0) {
    int cum = 0, b = NBINS - 1;
    for (; b >= 0; --b) { cum += (int)hist[b]; if (cum >= TOPN) break; }
    if (b < 0) b = 0;
    s_b1 = b;
    s_cum1 = cum - (int)hist[b];
  }
  __syncthreads();
  const int b1 = s_b1, cum1 = s_cum1;

  // ---------- Phase 2: refine with next 13 key bits (L2 pass, async-LDS) ---
  for (int i = tid; i < NBINS; i += T1) hist[i] = 0;
  __syncthreads();

  if (nChunk > 0) async_copy_b128(rowp4 + tid, &stage[0][tid]);
  for (int c = 0; c < nChunk; ++c) {
    if (c + 1 < nChunk)
      async_copy_b128(rowp4 + (size_t)(c + 1) * CHUNK_V4 + tid,
                      &stage[(c + 1) & 1][tid]);
    wait_async_le((c + 1 < nChunk) ? 1 : 0);     // chunk c done (in-order)
    __syncthreads();
    float4 v = stage[c & 1][tid];
    uint32_t kx = fkey(v.x); if ((int)(kx >> 19) == b1) atomicAdd(&hist[(kx >> 6) & (NBINS - 1)], 1u);
    uint32_t ky = fkey(v.y); if ((int)(ky >> 19) == b1) atomicAdd(&hist[(ky >> 6) & (NBINS - 1)], 1u);
    uint32_t kz = fkey(v.z); if ((int)(kz >> 19) == b1) atomicAdd(&hist[(kz >> 6) & (NBINS - 1)], 1u);
    uint32_t kw = fkey(v.w); if ((int)(kw >> 19) == b1) atomicAdd(&hist[(kw >> 6) & (NBINS - 1)], 1u);
    __syncthreads();
  }
  for (int i = nChunk * CHUNK_V4 + tid; i < nV4; i += T1) {
    float4 v = rowp4[i];
    uint32_t kx = fkey(v.x); if ((int)(kx >> 19) == b1) atomicAdd(&hist[(kx >> 6) & (NBINS - 1)], 1u);
    uint32_t ky = fkey(v.y); if ((int)(ky >> 19) == b1) atomicAdd(&hist[(ky >> 6) & (NBINS - 1)], 1u);
    uint32_t kz = fkey(v.z); if ((int)(kz >> 19) == b1) atomicAdd(&hist[(kz >> 6) & (NBINS - 1)], 1u);
    uint32_t kw = fkey(v.w); if ((int)(kw >> 19) == b1) atomicAdd(&hist[(kw >> 6) & (NBINS - 1)], 1u);
  }
  for (int i = (nV4 << 2) + tid; i < V; i += T1) {
    uint32_t k = fkey(rowp[i]);
    if ((int)(k >> 19) == b1) atomicAdd(&hist[(k >> 6) & (NBINS - 1)], 1u);
  }
  __syncthreads();
  if (tid == 0) {
    int cum = cum1, b = NBINS - 1;
    for (; b >= 0; --b) { cum += (int)hist[b]; if (cum >= TOPN) break; }
    if (b < 0) b = 0;
    s_thresh = ((uint32_t)b1 << 19) | ((uint32_t)b << 6);
    s_count = 0;
  }
  __syncthreads();
  const uint32_t thresh = s_thresh;

  // ---------- Phase 3: collect candidates (key >= thresh), ~TOPN+eps -------
  for (int i = tid; i < nV4; i += T1) {
    __builtin_prefetch(rowp4 + i + 4 * T1, 0, 3);   // global_prefetch ahead
    float4 v = rowp4[i];
    float  vv[4] = {v.x, v.y, v.z, v.w};
#pragma unroll
    for (int c = 0; c < 4; ++c) {
      if (fkey(vv[c]) >= thresh) {
        int pos = atomicAdd(&s_count, 1);
        if (pos < CAND_CAP) { cand_val[pos] = vv[c]; cand_idx[pos] = 4 * i + c; }
      }
    }
  }
  for (int i = (nV4 << 2) + tid; i < V; i += T1) {
    float x = rowp[i];
    if (fkey(x) >= thresh) {
      int pos = atomicAdd(&s_count, 1);
      if (pos < CAND_CAP) { cand_val[pos] = x; cand_idx[pos] = i; }
    }
  }
  __syncthreads();
  int n = s_count; if (n > CAND_CAP) n = CAND_CAP;
  for (int i = n + tid; i < CAND_CAP; i += T1) {
    cand_val[i] = -3.402823466e38f;   // rank lowest
    cand_idx[i] = 0x7fffffff;
  }
  __syncthreads();

  // ---------- Bitonic sort: value descending, index ascending (stable) -----
  for (int kk = 2; kk <= CAND_CAP; kk <<= 1) {
    for (int jj = kk >> 1; jj > 0; jj >>= 1) {
      for (int i = tid; i < CAND_CAP; i += T1) {
        int ixj = i ^ jj;
        if (ixj > i) {
          float va = cand_val[i], vb = cand_val[ixj];
          int   ia = cand_idx[i], ib = cand_idx[ixj];
          bool aFirst = (va > vb) || (va == vb && ia < ib);
          bool descRegion = ((i & kk) == 0);
          if (descRegion ? !aFirst : aFirst) {
            cand_val[i] = vb; cand_val[ixj] = va;
            cand_idx[i] = ib; cand_idx[ixj] = ia;
          }
        }
      }
      __syncthreads();
    }
  }

  for (int i = tid; i < TOPN; i += T1) {
    topv[(size_t)row * TOPN + i] = cand_val[i];
    topi[(size_t)row * TOPN + i] = cand_idx[i];
  }
}

// ---------------------------------------------------------------------------
// Finalize: per-row softmax / top-p / deterministic multinomial on <=100
// survivors. Also computes the global scalar g = min_b probs_cumsum[b, 0]
// (cumsum is monotone per row, so the global min is min over rows of p0).
// ---------------------------------------------------------------------------
__global__ __launch_bounds__(FT) void finalize_kernel(
    const float* __restrict__ topv, const int* __restrict__ topi,
    const float* __restrict__ params, int B, float* __restrict__ out) {
  __shared__ float smin[FT];
  const int tid = threadIdx.x;

  float lmin = 3.402823466e38f;
  for (int r = tid; r < B; r += FT) {
    int k = (int)params[r * 3 + 0]; if (k < 1) k = 1; if (k > TOPN) k = TOPN;
    float T = params[r * 3 + 2];
    const float* v = topv + (size_t)r * TOPN;
    float m = v[0] / T;
    float S = 0.f;
    for (int i = 0; i < k; ++i) S += expf(v[i] / T - m);
    lmin = fminf(lmin, 1.0f / S);   // p0 = exp(0)/S
  }
  smin[tid] = lmin;
  __syncthreads();
  for (int s = FT / 2; s > 0; s >>= 1) {
    if (tid < s) smin[tid] = fminf(smin[tid], smin[tid + s]);
    __syncthreads();
  }
  const float g = smin[0];
  __syncthreads();

  for (int r = tid; r < B; r += FT) {
    int k = (int)params[r * 3 + 0]; if (k < 1) k = 1; if (k > TOPN) k = TOPN;
    float tp = params[r * 3 + 1];
    float T  = params[r * 3 + 2];
    const float* v  = topv + (size_t)r * TOPN;
    const int*   id = topi + (size_t)r * TOPN;

    float m = v[0] / T;
    float S = 0.f;
    for (int i = 0; i < k; ++i) S += expf(v[i] / T - m);

    float teff = fmaxf(g, tp);

    // top-p mask on first-softmax cumsum; build second-softmax denominator.
    float c = 0.f, S2 = 0.f;
    for (int i = 0; i < k; ++i) {
      float e = expf(v[i] / T - m);
      c += e / S;
      bool masked = (i > 0) && (c > teff);
      if (!masked) S2 += e;
    }

    // counting pass: counts = sum(0.5 > cumsum2) over all positions
    // (masked positions contribute prob 0 but still occupy a rank slot).
    c = 0.f;
    float c2 = 0.f;
    int cnt = 0;
    for (int i = 0; i < k; ++i) {
      float e = expf(v[i] / T - m);
      c += e / S;
      bool masked = (i > 0) && (c > teff);
      if (!masked) c2 += e / S2;
      if (0.5f > c2) cnt++;
    }
    if (cnt > k - 1)    cnt = k - 1;
    if (cnt > TOPN - 1) cnt = TOPN - 1;
    out[r] = (float)id[cnt];
  }
}

extern "C" void kernel_launch(void* const* d_in, const int* in_sizes, int n_in,
                              void* d_out, int out_size, void* d_ws, size_t ws_size,
                              hipStream_t stream) {
  (void)n_in; (void)ws_size; (void)out_size;
  const float* logits = (const float*)d_in[0];   // [B, V] f32
  const float* params = (const float*)d_in[1];   // [B, 3] f32 (top_k, top_p, temp)
  const int B = in_sizes[1] / 3;
  const int V = in_sizes[0] / B;

  float* topv = (float*)d_ws;                          // [B, TOPN] f32
  int*   topi = (int*)(topv + (size_t)B * TOPN);       // [B, TOPN] i32
  float* out  = (float*)d_out;                         // [B] token ids as f32

  hipLaunchKernelGGL(topk_select_kernel, dim3(B), dim3(T1), 0, stream,
                     logits, V, topv, topi);
  hipLaunchKernelGGL(finalize_kernel, dim3(1), dim3(FT), 0, stream,
                     topv, topi, params, B, out);
}